// HGCN_64166811402452
// MI455X (gfx1250) — compile-verified
//
#include <hip/hip_runtime.h>
#include <hip/hip_bf16.h>

// HGCN hypergraph convolution for MI455X (gfx1250).
// N=8192 nodes, M=4096 edges, K=16 nodes/edge, D=128 features.
// Pipeline:
//   Z: zero accumulator + degree/slot counters
//   A: single streaming pass over dense H (128 MB -> ~5.5us @ 23.3 TB/s),
//      parallelized over 8 column-groups x 32 row-segments (256 blocks),
//      float4 loads; slots claimed with atomics (downstream use is
//      permutation-invariant, so index order is irrelevant)
//   B: dn[n] = 1/max(sqrt(deg), 1e-10)
//   C: per-edge block: gather 16x128 tile to LDS, Gram via
//      V_WMMA_F32_16X16X4_F32 (32 ops/edge), median of 120 pairwise
//      distances, w, atomic scatter
//   D: out = dn * acc

#define N_NODES 8192
#define M_EDGES 4096
#define K_E     16
#define D_F     128
#define EPS_F   1e-10f
#define N_SEG   32
#define ROWS_PER_SEG (N_NODES / N_SEG)   // 256

typedef __attribute__((ext_vector_type(2))) float v2f;
typedef __attribute__((ext_vector_type(8))) float v8f;

__global__ void hgcn_zero(float* __restrict__ acc, int* __restrict__ du,
                          int* __restrict__ ecnt) {
    int i = blockIdx.x * blockDim.x + threadIdx.x;
    if (i < N_NODES * D_F) acc[i] = 0.0f;
    if (i < N_NODES) du[i] = 0;
    if (i < M_EDGES) ecnt[i] = 0;
}

// Each thread owns 4 consecutive columns (float4 loads -> global_load_b128),
// each block scans one 256-row segment for a 512-column group. 256 blocks in
// flight so the 128 MB stream can actually run at HBM rate. Slot order within
// a column is arbitrary; all downstream consumers are permutation-invariant.
__global__ void __launch_bounds__(128) hgcn_scan_h(
    const float* __restrict__ H, int* __restrict__ edge_nodes,
    int* __restrict__ ecnt, int* __restrict__ du)
{
    const int m0 = blockIdx.x * 512 + threadIdx.x * 4;
    const int n0 = blockIdx.y * ROWS_PER_SEG;
    for (int n = n0; n < n0 + ROWS_PER_SEG; ++n) {
        if (n + 8 < N_NODES)
            __builtin_prefetch(&H[(size_t)(n + 8) * M_EDGES + m0], 0, 0);
        const float4 v = *(const float4*)&H[(size_t)n * M_EDGES + m0];
        if (v.x != 0.0f) {
            const int p = atomicAdd(&ecnt[m0 + 0], 1);
            if (p < K_E) edge_nodes[(m0 + 0) * K_E + p] = n;
            atomicAdd(&du[n], 1);
        }
        if (v.y != 0.0f) {
            const int p = atomicAdd(&ecnt[m0 + 1], 1);
            if (p < K_E) edge_nodes[(m0 + 1) * K_E + p] = n;
            atomicAdd(&du[n], 1);
        }
        if (v.z != 0.0f) {
            const int p = atomicAdd(&ecnt[m0 + 2], 1);
            if (p < K_E) edge_nodes[(m0 + 2) * K_E + p] = n;
            atomicAdd(&du[n], 1);
        }
        if (v.w != 0.0f) {
            const int p = atomicAdd(&ecnt[m0 + 3], 1);
            if (p < K_E) edge_nodes[(m0 + 3) * K_E + p] = n;
            atomicAdd(&du[n], 1);
        }
    }
}

__global__ void hgcn_dn(const int* __restrict__ du, float* __restrict__ dn) {
    int n = blockIdx.x * blockDim.x + threadIdx.x;
    if (n < N_NODES) dn[n] = 1.0f / fmaxf(sqrtf((float)du[n]), EPS_F);
}

__global__ void __launch_bounds__(128) hgcn_edge(
    const float* __restrict__ x, const int* __restrict__ edge_nodes,
    const float* __restrict__ dn, float* __restrict__ acc)
{
    // Row stride 132: bank stride 4 -> the 16 rows hit distinct LDS banks for
    // the WMMA fragment loads; 132*4 % 8 == 0 keeps float2 loads 8B-aligned.
    __shared__ float fsh[K_E][132];
    __shared__ float esum[D_F];
    __shared__ float gsh[K_E][K_E + 1];
    __shared__ float dsh[120];
    __shared__ int   nodes[K_E];
    __shared__ float dnk[K_E];
    __shared__ float sm[2];
    __shared__ float wsum;

    const int t = threadIdx.x;
    const int m = blockIdx.x;

    if (t < K_E) nodes[t] = edge_nodes[m * K_E + t];
    if (t == 0) wsum = 0.0f;
    __syncthreads();

    if (t < K_E) dnk[t] = dn[nodes[t]];
    // Gather 16 rows of x into LDS; iteration j loads row j with all 128
    // threads across d -> coalesced 512B per row.
    for (int i = t; i < K_E * D_F; i += 128) {
        const int k = i >> 7, d = i & 127;
        fsh[k][d] = x[(size_t)nodes[k] * D_F + d];
    }
    __syncthreads();

    // Edge aggregate: esum[d] = sum_k dn[n_k] * x[n_k][d]   (== (H^T (dn*x))[m])
    {
        float s = 0.0f;
        #pragma unroll
        for (int k = 0; k < K_E; ++k) s += dnk[k] * fsh[k][t];
        esum[t] = s;
    }
    __syncthreads();

    // Gram = f * f^T via V_WMMA_F32_16X16X4_F32 (fp32, K=4 per op, 32 ops for
    // D=128). For the symmetric product, the A fragment (16x4 MxK layout) and
    // the B fragment (4x16 KxN layout) are bit-identical, so pass A twice.
    if (t < 32) {   // EXEC all-ones within this wave, as WMMA requires
        const int r  = t & 15;
        const int hi = t >> 4;   // lanes 16..31 hold K-pair {2,3} of each chunk
        v8f c = {};
        #pragma unroll
        for (int kk = 0; kk < 32; ++kk) {
            const int d0 = kk * 4 + 2 * hi;
            const v2f a = *(const v2f*)&fsh[r][d0];
            c = __builtin_amdgcn_wmma_f32_16x16x4_f32(
                    false, a, false, a, (short)0, c, false, false);
        }
        // C/D layout: VGPR v, lane L -> (M = v + 8*(L>=16), N = L%16)
        #pragma unroll
        for (int v = 0; v < 8; ++v) gsh[v + 8 * hi][r] = c[v];
    }
    __syncthreads();

    // 120 upper-triangle squared distances
    float dv = 0.0f;
    if (t < 120) {
        int tt = t, rem = K_E - 1, ii = 0;
        while (tt >= rem) { tt -= rem; --rem; ++ii; }
        const int jj = ii + 1 + tt;
        dv = fmaxf(gsh[ii][ii] + gsh[jj][jj] - 2.0f * gsh[ii][jj], 0.0f);
        dsh[t] = dv;
    }
    __syncthreads();

    // median(120) = mean of order statistics 59 and 60; stable-rank selection
    if (t < 120) {
        int rank = 0;
        for (int u = 0; u < 120; ++u) {
            const float o = dsh[u];
            rank += (o < dv) || (o == dv && u < t);
        }
        if (rank == 59) sm[0] = dv;
        if (rank == 60) sm[1] = dv;
    }
    __syncthreads();

    const float sigma = 0.5f * (sm[0] + sm[1]);
    const float inv_s2 = 1.0f / (sigma * sigma);
    if (t < 120) atomicAdd(&wsum, __expf(-dv * inv_s2));
    __syncthreads();

    // w = wsum/(K*(K-1)); de = 1/K  (each column has exactly K ones)
    const float coef = wsum * (1.0f / (float)(K_E * (K_E - 1))) * (1.0f / (float)K_E);

    // Scatter H @ (w*de*esum) into node accumulator
    for (int i = t; i < K_E * D_F; i += 128) {
        const int k = i >> 7, d = i & 127;
        atomicAdd(&acc[(size_t)nodes[k] * D_F + d], coef * esum[d]);
    }
}

__global__ void hgcn_finalize(const float* __restrict__ acc,
                              const float* __restrict__ dn,
                              float* __restrict__ out)
{
    int i = blockIdx.x * blockDim.x + threadIdx.x;
    if (i < N_NODES * D_F) out[i] = dn[i >> 7] * acc[i];
}

extern "C" void kernel_launch(void* const* d_in, const int* in_sizes, int n_in,
                              void* d_out, int out_size, void* d_ws, size_t ws_size,
                              hipStream_t stream) {
    (void)in_sizes; (void)n_in; (void)out_size; (void)ws_size;

    const float* x = (const float*)d_in[0];           // (N, D) f32
    const float* H = (const float*)d_in[1];           // (N, M) f32
    float* out = (float*)d_out;                       // (N, D) f32

    // Workspace partition (~4.5 MB total)
    char* ws = (char*)d_ws;
    size_t off = 0;
    float* acc = (float*)(ws + off); off += (size_t)N_NODES * D_F * 4;  // 4 MB
    int*   du  = (int*)  (ws + off); off += (size_t)N_NODES * 4;        // 32 KB
    float* dnv = (float*)(ws + off); off += (size_t)N_NODES * 4;        // 32 KB
    int*   en  = (int*)  (ws + off); off += (size_t)M_EDGES * K_E * 4;  // 256 KB
    int*   ecnt= (int*)  (ws + off); off += (size_t)M_EDGES * 4;        // 16 KB

    hgcn_zero<<<(N_NODES * D_F + 255) / 256, 256, 0, stream>>>(acc, du, ecnt);
    hgcn_scan_h<<<dim3(M_EDGES / 512, N_SEG), 128, 0, stream>>>(H, en, ecnt, du);
    hgcn_dn<<<(N_NODES + 255) / 256, 256, 0, stream>>>(du, dnv);
    hgcn_edge<<<M_EDGES, 128, 0, stream>>>(x, en, dnv, acc);
    hgcn_finalize<<<(N_NODES * D_F + 255) / 256, 256, 0, stream>>>(acc, dnv, out);
}